// My_GATLayer_23862838297009
// MI455X (gfx1250) — compile-verified
//
#include <hip/hip_runtime.h>
#include <hip/hip_bf16.h>
#include <math.h>

#define N_NODES 10000
#define N_EDGES 160000
#define FDIM 256
#define EDIM 64
#define NEG_SLOPE 0.01f
#define LDS_STRIDE 260  // 256 + 4 pad: lane m hits bank 4*m -> conflict-free b64 reads

typedef __attribute__((ext_vector_type(2))) float v2f;
typedef __attribute__((ext_vector_type(8))) float v8f;

// ---------------------------------------------------------------------------
// Kernel 1: init scratch (agg=0, m=-inf, denom=0)
// ---------------------------------------------------------------------------
__global__ __launch_bounds__(256) void init_kernel(float* __restrict__ agg,
                                                   float* __restrict__ m_buf,
                                                   float* __restrict__ denom) {
    int i = blockIdx.x * 256 + threadIdx.x;
    if (i < N_NODES * FDIM) agg[i] = 0.0f;
    if (i < N_NODES) { m_buf[i] = -INFINITY; denom[i] = 0.0f; }
}

// ---------------------------------------------------------------------------
// Kernel 2: dual GEMM  hs = h@Wself^T, z = h@Wfunc^T  via V_WMMA_F32_16X16X4_F32
// One block (8 waves) per 16-row panel of h; panel staged in LDS via
// GLOBAL_LOAD_ASYNC_TO_LDS_B128 (ASYNCcnt). Each wave keeps 4 accumulators so
// one LDS A-load feeds 4 independent WMMAs per K-step (4x A reuse, dense XDL
// issue). Waves 0-3 -> W_self col-tiles 0..15, waves 4-7 -> W_func.
// ---------------------------------------------------------------------------
__global__ __launch_bounds__(256) void gemm_dual_kernel(const float* __restrict__ h,
                                                        const float* __restrict__ Wself,
                                                        const float* __restrict__ Wfunc,
                                                        float* __restrict__ hs,
                                                        float* __restrict__ z) {
    __shared__ float lds[16 * LDS_STRIDE];
    const int tid = threadIdx.x;
    const int rowBase = blockIdx.x * 16;

    // --- Async-stage A panel h[rowBase:rowBase+16][0:256] -> LDS (16 KB) ---
    // GVS mode: uniform SGPR base, per-lane 32-bit byte offset, 16 B per issue.
    const float* srcp = h + (size_t)rowBase * FDIM;
#pragma unroll
    for (int i = 0; i < 4; ++i) {
        int f4 = tid + 256 * i;          // 0..1023 16B-chunks
        int r  = f4 >> 6;                // row (64 chunks per row)
        int c4 = f4 & 63;
        unsigned ldsAddr = (unsigned)(size_t)&lds[r * LDS_STRIDE + c4 * 4];
        unsigned voff    = (unsigned)(f4 * 16);
        asm volatile("global_load_async_to_lds_b128 %0, %1, %2 offset:0"
                     :: "v"(ldsAddr), "v"(voff), "s"(srcp)
                     : "memory");
    }
    asm volatile("s_wait_asynccnt 0x0" ::: "memory");
    __syncthreads();

    const int wave = tid >> 5;
    const int lane = tid & 31;
    const int half = lane >> 4;          // K-pair select per WMMA A/B layout
    const int l15  = lane & 15;          // M (for A) / N (for B)

    const float* Wp  = (wave >= 4) ? Wfunc : Wself;
    const int ct0    = (wave & 3) * 4;   // first of 4 col-tiles for this wave
    // B[k][n] = W[n][k]: per-tile row pointers (columns of W, L2-resident)
    const float* wr0 = Wp + (size_t)((ct0 + 0) * 16 + l15) * FDIM;
    const float* wr1 = Wp + (size_t)((ct0 + 1) * 16 + l15) * FDIM;
    const float* wr2 = Wp + (size_t)((ct0 + 2) * 16 + l15) * FDIM;
    const float* wr3 = Wp + (size_t)((ct0 + 3) * 16 + l15) * FDIM;

    v8f c0 = {}, c1 = {}, c2 = {}, c3 = {};
#pragma unroll 2
    for (int k = 0; k < FDIM; k += 4) {
        // A 16x4: lanes 0-15 K={k,k+1}, lanes 16-31 K={k+2,k+3}, M = l15
        v2f a  = *(const v2f*)&lds[l15 * LDS_STRIDE + k + 2 * half];
        v2f b0 = *(const v2f*)&wr0[k + 2 * half];
        v2f b1 = *(const v2f*)&wr1[k + 2 * half];
        v2f b2 = *(const v2f*)&wr2[k + 2 * half];
        v2f b3 = *(const v2f*)&wr3[k + 2 * half];
        c0 = __builtin_amdgcn_wmma_f32_16x16x4_f32(false, a, false, b0, (short)0, c0, false, false);
        c1 = __builtin_amdgcn_wmma_f32_16x16x4_f32(false, a, false, b1, (short)0, c1, false, false);
        c2 = __builtin_amdgcn_wmma_f32_16x16x4_f32(false, a, false, b2, (short)0, c2, false, false);
        c3 = __builtin_amdgcn_wmma_f32_16x16x4_f32(false, a, false, b3, (short)0, c3, false, false);
    }

    float* outp = (wave >= 4) ? z : hs;
    // C/D layout: lanes 0-15 -> M=r, lanes 16-31 -> M=8+r
    const size_t rbase = (size_t)(rowBase + 8 * half) * FDIM;
#pragma unroll
    for (int r = 0; r < 8; ++r) {
        size_t rowOff = rbase + (size_t)r * FDIM;
        outp[rowOff + (ct0 + 0) * 16 + l15] = c0[r];
        outp[rowOff + (ct0 + 1) * 16 + l15] = c1[r];
        outp[rowOff + (ct0 + 2) * 16 + l15] = c2[r];
        outp[rowOff + (ct0 + 3) * 16 + l15] = c3[r];
    }
}

// ---------------------------------------------------------------------------
// Kernel 3: per-node attention partials s1[n]=a1.z[n], s2[n]=a2.z[n]
// One wave32 per node.
// ---------------------------------------------------------------------------
__global__ __launch_bounds__(256) void node_score_kernel(const float* __restrict__ z,
                                                         const float* __restrict__ Watt,
                                                         float* __restrict__ s1,
                                                         float* __restrict__ s2) {
    int node = blockIdx.x * 8 + (threadIdx.x >> 5);
    int lane = threadIdx.x & 31;
    if (node >= N_NODES) return;
    const float* zr = z + (size_t)node * FDIM;
    float a = 0.0f, b = 0.0f;
#pragma unroll
    for (int k = lane; k < FDIM; k += 32) {
        float zv = zr[k];
        a += zv * Watt[k];
        b += zv * Watt[FDIM + k];
    }
#pragma unroll
    for (int off = 16; off; off >>= 1) {
        a += __shfl_down(a, off, 32);
        b += __shfl_down(b, off, 32);
    }
    if (lane == 0) { s1[node] = a; s2[node] = b; }
}

// ---------------------------------------------------------------------------
// Kernel 4: per-edge score e = leakyrelu(s1[src]+s2[dst]+a3.w)  +  segment max
// One wave32 per edge. Float atomic max via monotonic int/uint punning.
// ---------------------------------------------------------------------------
__global__ __launch_bounds__(256) void edge_score_kernel(const float* __restrict__ w,
                                                         const int* __restrict__ esrc,
                                                         const int* __restrict__ edst,
                                                         const float* __restrict__ s1,
                                                         const float* __restrict__ s2,
                                                         const float* __restrict__ Watt,
                                                         float* __restrict__ e_buf,
                                                         float* __restrict__ m_buf) {
    int edge = blockIdx.x * 8 + (threadIdx.x >> 5);
    int lane = threadIdx.x & 31;
    if (edge >= N_EDGES) return;
    const float* wr = w + (size_t)edge * EDIM;
    const float* a3 = Watt + 2 * FDIM;
    float p = wr[lane] * a3[lane] + wr[lane + 32] * a3[lane + 32];
#pragma unroll
    for (int off = 16; off; off >>= 1) p += __shfl_down(p, off, 32);
    if (lane == 0) {
        int s = esrc[edge], d = edst[edge];
        float x  = s1[s] + s2[d] + p;
        float ev = x > 0.0f ? x : NEG_SLOPE * x;
        e_buf[edge] = ev;
        if (ev >= 0.0f) atomicMax((int*)&m_buf[d], __float_as_int(ev));
        else            atomicMin((unsigned int*)&m_buf[d], __float_as_uint(ev));
    }
}

// ---------------------------------------------------------------------------
// Kernel 5: e_exp = exp(e - m[dst]); denom[dst] += e_exp
// ---------------------------------------------------------------------------
__global__ __launch_bounds__(256) void exp_denom_kernel(const int* __restrict__ edst,
                                                        float* __restrict__ e_buf,
                                                        const float* __restrict__ m_buf,
                                                        float* __restrict__ denom) {
    int i = blockIdx.x * 256 + threadIdx.x;
    if (i >= N_EDGES) return;
    int d = edst[i];
    float ee = expf(e_buf[i] - m_buf[d]);
    e_buf[i] = ee;
    atomicAdd(&denom[d], ee);
}

// ---------------------------------------------------------------------------
// Kernel 6: agg[dst] += (e_exp/denom[dst]) * z[src]   (one block per edge)
// Memory-bound hot loop: coalesced 1KB gather + 1KB f32 atomic-add per edge.
// ---------------------------------------------------------------------------
__global__ __launch_bounds__(256) void scatter_kernel(const int* __restrict__ esrc,
                                                      const int* __restrict__ edst,
                                                      const float* __restrict__ e_buf,
                                                      const float* __restrict__ denom,
                                                      const float* __restrict__ z,
                                                      float* __restrict__ agg) {
    int edge = blockIdx.x;
    int f    = threadIdx.x;
    int s = esrc[edge], d = edst[edge];
    float alpha = e_buf[edge] / denom[d];       // broadcast loads (same addr per block)
    atomicAdd(&agg[(size_t)d * FDIM + f], alpha * z[(size_t)s * FDIM + f]);
}

// ---------------------------------------------------------------------------
// Kernel 7: out = h + relu( deg>0 ? hs+agg : h ),  deg>0 <=> m > -inf
// ---------------------------------------------------------------------------
__global__ __launch_bounds__(256) void final_kernel(const float* __restrict__ h,
                                                    const float* __restrict__ hs,
                                                    const float* __restrict__ agg,
                                                    const float* __restrict__ m_buf,
                                                    float* __restrict__ out) {
    int i = blockIdx.x * 256 + threadIdx.x;
    if (i >= N_NODES * FDIM) return;
    int n = i >> 8;
    float hv = h[i];
    float hn = (m_buf[n] > -INFINITY) ? (hs[i] + agg[i]) : hv;
    hn = hn > 0.0f ? hn : 0.0f;
    out[i] = hv + hn;
}

// ---------------------------------------------------------------------------
extern "C" void kernel_launch(void* const* d_in, const int* in_sizes, int n_in,
                              void* d_out, int out_size, void* d_ws, size_t ws_size,
                              hipStream_t stream) {
    const float* h      = (const float*)d_in[0];
    const float* w      = (const float*)d_in[1];
    // d_in[2] = snorm_n (unused by the reference computation)
    const int*   esrc   = (const int*)d_in[3];
    const int*   edst   = (const int*)d_in[4];
    const float* Wself  = (const float*)d_in[5];
    const float* Wfunc  = (const float*)d_in[6];
    const float* Watt   = (const float*)d_in[7];
    float*       out    = (float*)d_out;

    // Workspace layout (floats): ~31.5 MB total
    float* ws  = (float*)d_ws;
    float* hs  = ws;                            // N*F
    float* z   = hs  + (size_t)N_NODES * FDIM;  // N*F
    float* agg = z   + (size_t)N_NODES * FDIM;  // N*F
    float* s1  = agg + (size_t)N_NODES * FDIM;  // N
    float* s2  = s1  + N_NODES;                 // N
    float* mb  = s2  + N_NODES;                 // N
    float* dn  = mb  + N_NODES;                 // N
    float* eb  = dn  + N_NODES;                 // E

    init_kernel<<<(N_NODES * FDIM + 255) / 256, 256, 0, stream>>>(agg, mb, dn);
    gemm_dual_kernel<<<N_NODES / 16, 256, 0, stream>>>(h, Wself, Wfunc, hs, z);
    node_score_kernel<<<(N_NODES + 7) / 8, 256, 0, stream>>>(z, Watt, s1, s2);
    edge_score_kernel<<<(N_EDGES + 7) / 8, 256, 0, stream>>>(w, esrc, edst, s1, s2, Watt, eb, mb);
    exp_denom_kernel<<<(N_EDGES + 255) / 256, 256, 0, stream>>>(edst, eb, mb, dn);
    scatter_kernel<<<N_EDGES, 256, 0, stream>>>(esrc, edst, eb, dn, z, agg);
    final_kernel<<<(N_NODES * FDIM + 255) / 256, 256, 0, stream>>>(h, hs, agg, mb, out);
}